// DataEmbedding_11484742550040
// MI455X (gfx1250) — compile-verified
//
#include <hip/hip_runtime.h>

typedef float v2f __attribute__((ext_vector_type(2)));
typedef float v8f __attribute__((ext_vector_type(8)));

#define B_    64
#define SEQ_  1534
#define CIN_  21
#define DM_   516
#define KS_   8
#define NK_   74
#define TAU_  3
#define PAD_  21
#define LDSX  1568   // >= SEQ_+PAD_ = 1555
#define LDSK  640    // 80 rows x 8, zero-padded past NK_=74
#define NROW  80     // 5 m-tiles * 16 rows (padded)

__global__ __launch_bounds__(256)
void data_embed_kernel(const float* __restrict__ x,
                       const int*   __restrict__ xmark,
                       const float* __restrict__ kern,
                       const float* __restrict__ pe,
                       const float* __restrict__ hour_tab,
                       const float* __restrict__ weekday_tab,
                       const float* __restrict__ day_tab,
                       const float* __restrict__ month_tab,
                       float* __restrict__ out)
{
    __shared__ float lx[LDSX];
    __shared__ float lk[LDSK];
    __shared__ int4  lidx[NROW];   // (month, day, weekday, hour) per output row

    const int blk = blockIdx.x;
    const int b   = blk / CIN_;
    const int ch  = blk % CIN_;
    const int tid = threadIdx.x;

    const unsigned outB = (unsigned)b * (SEQ_ * DM_);   // < 2^31, 32-bit safe
    const int      xmB  = b * (SEQ_ * 5);

    // Stage zero-padded x[b, :, ch] into LDS (front pad of 21 zeros).
    for (int p = tid; p < LDSX; p += 256) {
        float v = 0.0f;
        int s = p - PAD_;
        if (s >= 0 && s < SEQ_)
            v = x[((size_t)b * SEQ_ + s) * CIN_ + ch];
        lx[p] = v;
    }
    // Stage kernels, zero-padded to 80 rows so padded A-tile rows contribute 0.
    for (int t = tid; t < LDSK; t += 256)
        lk[t] = (t < NK_ * KS_) ? kern[t] : 0.0f;
    // Stage time-mark indices for this block's 74 rows (one aligned int4/row).
    for (int t = tid; t < NROW * 4; t += 256) {
        int n = t >> 2, j = t & 3;
        int f = n * CIN_ + ch;
        int v = 0;
        if (f < SEQ_) v = xmark[xmB + f * 5 + j];
        ((int*)lidx)[t] = v;
    }
    __syncthreads();

    const int wave = tid >> 5;     // wave32
    const int lane = tid & 31;
    const int nlo  = lane & 15;
    const int hi   = lane >> 4;    // lane half selects K pair (f32 16x16x4 layout)
    const int kofs = 2 * hi;

    // A fragments: 5 M-tiles x 2 K-steps; A[M][k] = kernels[mt*16+M][kk*4+k]
    v2f afrag[5][2];
    #pragma unroll
    for (int mt = 0; mt < 5; ++mt) {
        int n = mt * 16 + nlo;     // M = lane&15
        #pragma unroll
        for (int kk = 0; kk < 2; ++kk) {
            afrag[mt][kk].x = lk[n * KS_ + kk * 4 + kofs];
            afrag[mt][kk].y = lk[n * KS_ + kk * 4 + kofs + 1];
        }
    }

    #pragma unroll
    for (int mt = 0; mt < 5; ++mt) {
        // Per-lane output rows of this M-tile: M = r + 8*hi  (C/D layout).
        // Precompute 32-bit row offsets so every gather/store is base[off + d].
        bool     ok[8];
        int      pOff[8], hOff[8], wOff[8], dOff[8], mOff[8];
        unsigned oOff[8];
        #pragma unroll
        for (int r = 0; r < 8; ++r) {
            int n = mt * 16 + r + 8 * hi;
            int f = n * CIN_ + ch;
            ok[r]   = (n < 73) || (f < SEQ_);          // compile-time true for mt<4
            int fc  = ok[r] ? f : (SEQ_ - 1);          // clamp so batched loads stay in-bounds
            pOff[r] = fc * DM_;
            oOff[r] = outB + (unsigned)f * DM_;        // store is guarded by ok[r]
            int4 q  = lidx[n];
            mOff[r] = q.x * DM_;
            dOff[r] = q.y * DM_;
            wOff[r] = q.z * DM_;
            hOff[r] = q.w * DM_;
        }

        for (int nt = wave; nt < 33; nt += 8) {   // 33 N-tiles over D_MODEL=516
            int  w0 = nt * 16;
            int  d  = w0 + nlo;                   // N = lane&15 -> output column
            bool dv = (d < DM_);
            int  dc = dv ? d : (DM_ - 1);         // clamp: garbage cols never stored

            // B[k][N] = lx[d*3 + kk*4 + k]; lane holds k = kk*4 + 2*hi + {0,1}
            int p0 = dc * TAU_ + kofs;
            v2f bf0, bf1;
            bf0.x = lx[p0];
            bf0.y = lx[p0 + 1];
            bf1.x = lx[p0 + 4];
            bf1.y = lx[p0 + 5];

            v8f c = {};
            c = __builtin_amdgcn_wmma_f32_16x16x4_f32(false, afrag[mt][0], false, bf0,
                                                      (short)0, c, false, false);
            c = __builtin_amdgcn_wmma_f32_16x16x4_f32(false, afrag[mt][1], false, bf1,
                                                      (short)0, c, false, false);

            if (dv) {
                // Phase 1: issue ALL gathers into independent regs (clauses, one wait).
                float vpe[8], vh[8], vw[8], vd[8], vm[8];
                #pragma unroll
                for (int r = 0; r < 8; ++r) vpe[r] = pe[pOff[r] + d];
                #pragma unroll
                for (int r = 0; r < 8; ++r) vh[r]  = hour_tab[hOff[r] + d];
                #pragma unroll
                for (int r = 0; r < 8; ++r) vw[r]  = weekday_tab[wOff[r] + d];
                #pragma unroll
                for (int r = 0; r < 8; ++r) vd[r]  = day_tab[dOff[r] + d];
                #pragma unroll
                for (int r = 0; r < 8; ++r) vm[r]  = month_tab[mOff[r] + d];

                // Phase 2: sum + NT store.
                #pragma unroll
                for (int r = 0; r < 8; ++r) {
                    if (ok[r]) {
                        float v = c[r] + vpe[r] + vh[r] + vw[r] + vd[r] + vm[r];
                        __builtin_nontemporal_store(v, &out[oOff[r] + (unsigned)d]);
                    }
                }
            }
        }
    }
}

extern "C" void kernel_launch(void* const* d_in, const int* in_sizes, int n_in,
                              void* d_out, int out_size, void* d_ws, size_t ws_size,
                              hipStream_t stream) {
    const float* x           = (const float*)d_in[0];
    const int*   xm          = (const int*)  d_in[1];
    const float* kern        = (const float*)d_in[2];
    const float* pe          = (const float*)d_in[3];
    const float* hour_tab    = (const float*)d_in[4];
    const float* weekday_tab = (const float*)d_in[5];
    const float* day_tab     = (const float*)d_in[6];
    const float* month_tab   = (const float*)d_in[7];
    float* out = (float*)d_out;

    dim3 grid(B_ * CIN_);   // one workgroup per (batch, channel)
    data_embed_kernel<<<grid, 256, 0, stream>>>(x, xm, kern, pe, hour_tab,
                                                weekday_tab, day_tab, month_tab, out);
}